// oursgat_60198261620972
// MI455X (gfx1250) — compile-verified
//
#include <hip/hip_runtime.h>

typedef __attribute__((ext_vector_type(2))) float v2f;
typedef __attribute__((ext_vector_type(8))) float v8f;
typedef __attribute__((ext_vector_type(4))) int   v4i;

typedef v4i __attribute__((address_space(1)))* as1_v4i_ptr;  // global
typedef v4i __attribute__((address_space(3)))* as3_v4i_ptr;  // LDS

#define LN_EPS 1e-5f
#define EPS_F  1e-10f

#if defined(__has_builtin)
#if __has_builtin(__builtin_amdgcn_global_load_async_to_lds_b128)
#define HAS_ASYNC_LDS 1
#endif
#endif
#ifndef HAS_ASYNC_LDS
#define HAS_ASYNC_LDS 0
#endif

// ---------------------------------------------------------------------------
// Kernel 1: V = x@W1+b1 ; Q = relu(x@W2+b2); h = (sum(layer_norm(Q),axis=1))^2
// fp32 WMMA (V_WMMA_F32_16X16X4_F32). Block = 128 threads = 4 waves,
// each wave owns 16 rows; block covers 64 rows. K = 64 in steps of 4.
// LDS layouts (all row stride 68 floats = 34 v2f -> conflict-free b64 reads):
//   sx2   [row][k]  : A frag  a = sx2[row*34 + ka/2]      (ka = k0+2*half)
//   swXt2 [col][k]  : B frag  b = swXt2[col*34 + ka/2]
// A frag (16x4 f32): a[v] = A[m][2*half+v]
// B frag (4x16 f32): b[v] = B[2*half+v][m]
// C (16x16 f32):     c[v] = C[v+8*half][m]
// ---------------------------------------------------------------------------
__global__ __launch_bounds__(128) void gemm_h_kernel(
    const float* __restrict__ x,
    const float* __restrict__ W1, const float* __restrict__ b1,
    const float* __restrict__ W2, const float* __restrict__ b2,
    float* __restrict__ V, float* __restrict__ hbuf, int N)
{
    __shared__ v2f sx2[64 * 34];    // x tile, [row][k], stride 68 floats
    __shared__ v2f sw1t2[64 * 34];  // W1^T,   [col][k], stride 68 floats
    __shared__ v2f sw2t2[64 * 34];  // W2^T,   [col][k], stride 68 floats
    __shared__ float sb1[64];
    __shared__ float sb2[64];

    float* sxf   = (float*)sx2;
    float* sw1tf = (float*)sw1t2;
    float* sw2tf = (float*)sw2t2;

    const int tid  = threadIdx.x;
    const int row0 = blockIdx.x * 64;
    const bool fulltile = (row0 + 64 <= N);

#if HAS_ASYNC_LDS
    if (fulltile) {
        // per-lane-addressed async copy: 128 lanes x b128 x 8 iters = 64x64 f32
#pragma unroll
        for (int it = 0; it < 8; ++it) {
            const int j  = it * 128 + tid;
            const int r  = j >> 4;       // 16 float4 per row
            const int c4 = (j & 15) * 4;
            v4i* gpv = (v4i*)(x + (size_t)(row0 + r) * 64 + c4);
            v4i* lpv = (v4i*)&sxf[r * 68 + c4];
            __builtin_amdgcn_global_load_async_to_lds_b128(
                (as1_v4i_ptr)gpv, (as3_v4i_ptr)lpv, 0, 0);
        }
    } else
#endif
    {
        for (int i = tid; i < 4096; i += 128) {
            const int r = i >> 6, c = i & 63;
            sxf[r * 68 + c] = (row0 + r < N) ? x[(size_t)(row0 + r) * 64 + c] : 0.0f;
        }
    }

    // W1/W2 transposed into LDS ([col][k], stride 68)
    for (int i = tid; i < 4096; i += 128) {
        const int r = i >> 6, c = i & 63;
        sw1tf[c * 68 + r] = W1[i];
        sw2tf[c * 68 + r] = W2[i];
    }
    if (tid < 64) { sb1[tid] = b1[tid]; sb2[tid] = b2[tid]; }

#if HAS_ASYNC_LDS
    if (fulltile) { asm volatile("s_wait_asynccnt 0x0" ::: "memory"); }
#endif
    __syncthreads();

    const int lane = tid & 31;
    const int wid  = tid >> 5;
    const int m    = lane & 15;
    const int half = lane >> 4;

    v8f accV[4], accQ[4];
#pragma unroll
    for (int nt = 0; nt < 4; ++nt) {
        accV[nt] = (v8f){0,0,0,0,0,0,0,0};
        accQ[nt] = (v8f){0,0,0,0,0,0,0,0};
    }

    const int arow = (wid * 16 + m) * 34;
#pragma unroll 4
    for (int k0 = 0; k0 < 64; k0 += 4) {
        const int kh = (k0 + 2 * half) >> 1;   // v2f index; k0+2*half is even
        const v2f a = sx2[arow + kh];
#pragma unroll
        for (int nt = 0; nt < 4; ++nt) {
            const int cb = (nt * 16 + m) * 34 + kh;
            accV[nt] = __builtin_amdgcn_wmma_f32_16x16x4_f32(
                false, a, false, sw1t2[cb], (short)0, accV[nt], false, false);
            accQ[nt] = __builtin_amdgcn_wmma_f32_16x16x4_f32(
                false, a, false, sw2t2[cb], (short)0, accQ[nt], false, false);
        }
    }

    // ---- V = acc + b1, store ----
#pragma unroll
    for (int nt = 0; nt < 4; ++nt) {
        const int col = nt * 16 + m;
        const float bias = sb1[col];
#pragma unroll
        for (int v = 0; v < 8; ++v) {
            const int row = row0 + wid * 16 + v + 8 * half;
            if (row < N) V[(size_t)row * 64 + col] = accV[nt][v] + bias;
        }
    }

    // ---- Q = relu(acc + b2); per-row layernorm; h = (sum of normed)^2 ----
#pragma unroll
    for (int v = 0; v < 8; ++v) {
        float q[4];
        float s1 = 0.0f, s2 = 0.0f;
#pragma unroll
        for (int nt = 0; nt < 4; ++nt) {
            const int col = nt * 16 + m;
            q[nt] = fmaxf(accQ[nt][v] + sb2[col], 0.0f);
            s1 += q[nt];
            s2 += q[nt] * q[nt];
        }
        // row lives across 16 lanes of one half -> width-16 xor reduce
#pragma unroll
        for (int mk = 1; mk < 16; mk <<= 1) {
            s1 += __shfl_xor(s1, mk, 16);
            s2 += __shfl_xor(s2, mk, 16);
        }
        const float mu  = s1 * (1.0f / 64.0f);
        const float var = s2 * (1.0f / 64.0f) - mu * mu;
        const float rs  = rsqrtf(var + LN_EPS);
        float tsum = 0.0f;
#pragma unroll
        for (int nt = 0; nt < 4; ++nt) tsum += (q[nt] - mu) * rs;
#pragma unroll
        for (int mk = 1; mk < 16; mk <<= 1) tsum += __shfl_xor(tsum, mk, 16);

        if (m == 0) {
            const int row = row0 + wid * 16 + v + 8 * half;
            if (row < N) hbuf[row] = tsum * tsum;
        }
    }
}

// ---------------------------------------------------------------------------
// Kernel 0: init min/max sentinels (h >= 0 -> uint bit ordering == float order)
// ---------------------------------------------------------------------------
__global__ void init_mm_kernel(unsigned* mm)
{
    mm[0] = 0u;           // running max (>= 0)
    mm[1] = 0x7F7FFFFFu;  // running min (FLT_MAX bits)
}

// ---------------------------------------------------------------------------
// Kernel 2: max/min of h[src[e]] over all edges
// ---------------------------------------------------------------------------
__global__ __launch_bounds__(256) void minmax_kernel(
    const float* __restrict__ h, const int* __restrict__ src,
    int E, unsigned* __restrict__ mm)
{
    const int t = blockIdx.x * 256 + threadIdx.x;
    float vmax = 0.0f;
    float vmin = 3.402823466e38f;
    if (t < E) {
        const float v = h[src[t]];
        vmax = v; vmin = v;
    }
#pragma unroll
    for (int mk = 1; mk < 32; mk <<= 1) {
        vmax = fmaxf(vmax, __shfl_xor(vmax, mk, 32));
        vmin = fminf(vmin, __shfl_xor(vmin, mk, 32));
    }
    __shared__ float smax[8], smin[8];
    const int lane = threadIdx.x & 31, wid = threadIdx.x >> 5;
    if (lane == 0) { smax[wid] = vmax; smin[wid] = vmin; }
    __syncthreads();
    if (threadIdx.x == 0) {
        float bmax = smax[0], bmin = smin[0];
#pragma unroll
        for (int i = 1; i < 8; ++i) {
            bmax = fmaxf(bmax, smax[i]);
            bmin = fminf(bmin, smin[i]);
        }
        atomicMax(mm + 0, __float_as_uint(bmax));
        atomicMin(mm + 1, __float_as_uint(bmin));
    }
}

// ---------------------------------------------------------------------------
// Kernel 3: attention[e] = cos(pi/(2m+eps)*(h_src-h_dst)); rowsum[src] += att
// ---------------------------------------------------------------------------
__global__ __launch_bounds__(256) void att_kernel(
    const float* __restrict__ h, const int* __restrict__ src,
    const int* __restrict__ dst, const unsigned* __restrict__ mm,
    float* __restrict__ att, float* __restrict__ rowsum, int E)
{
    const int e = blockIdx.x * 256 + threadIdx.x;
    if (e >= E) return;
    const float mrange = __uint_as_float(mm[0]) - __uint_as_float(mm[1]);
    const float c = 3.14159265358979323846f / (2.0f * mrange + EPS_F);
    const float a = cosf(c * (h[src[e]] - h[dst[e]]));
    att[e] = a;
    atomicAdd(rowsum + src[e], a);
}

// ---------------------------------------------------------------------------
// Kernel 4: out[src[e]] += (att[e]/rowsum[src[e]] + eps) * V[dst[e]]
// thread = (edge, column) -> 64 consecutive lanes read one V row coalesced
// ---------------------------------------------------------------------------
__global__ __launch_bounds__(256) void scatter_kernel(
    const float* __restrict__ V, const int* __restrict__ src,
    const int* __restrict__ dst, const float* __restrict__ att,
    const float* __restrict__ rowsum, float* __restrict__ out, int E)
{
    const unsigned t = blockIdx.x * 256u + threadIdx.x;
    if (t >= (unsigned)E * 64u) return;
    const int e = (int)(t >> 6);
    const int c = (int)(t & 63u);
    const int s = src[e];
    const int d = dst[e];
    const float w = att[e] / rowsum[s] + EPS_F;
    atomicAdd(out + (size_t)s * 64 + c, w * V[(size_t)d * 64 + c]);
}

// ---------------------------------------------------------------------------
extern "C" void kernel_launch(void* const* d_in, const int* in_sizes, int n_in,
                              void* d_out, int out_size, void* d_ws, size_t ws_size,
                              hipStream_t stream)
{
    const float* x  = (const float*)d_in[0];
    const int*  ei  = (const int*)d_in[1];   // int32 (JAX x64 disabled)
    const float* W1 = (const float*)d_in[2];
    const float* b1 = (const float*)d_in[3];
    const float* W2 = (const float*)d_in[4];
    const float* b2 = (const float*)d_in[5];

    const int N = in_sizes[0] / 64;
    const int E = in_sizes[1] / 2;
    const int* src = ei;       // edge_index[0]
    const int* dst = ei + E;   // edge_index[1]

    float* ws     = (float*)d_ws;
    float* V      = ws;                        // N*64
    float* h      = V + (size_t)N * 64;        // N
    float* rowsum = h + N;                     // N
    float* att    = rowsum + N;                // E
    unsigned* mm  = (unsigned*)(att + E);      // 2

    float* out = (float*)d_out;

    (void)hipMemsetAsync(out, 0, (size_t)N * 64 * sizeof(float), stream);
    (void)hipMemsetAsync(rowsum, 0, (size_t)N * sizeof(float), stream);
    init_mm_kernel<<<1, 1, 0, stream>>>(mm);

    gemm_h_kernel<<<(N + 63) / 64, 128, 0, stream>>>(x, W1, b1, W2, b2, V, h, N);
    minmax_kernel<<<(E + 255) / 256, 256, 0, stream>>>(h, src, E, mm);
    att_kernel<<<(E + 255) / 256, 256, 0, stream>>>(h, src, dst, mm, att, rowsum, E);

    const unsigned total = (unsigned)E * 64u;
    scatter_kernel<<<(total + 255u) / 256u, 256, 0, stream>>>(V, src, dst, att, rowsum, out, E);
}